// AGNNConv_70789650972793
// MI455X (gfx1250) — compile-verified
//
#include <hip/hip_runtime.h>

typedef float v2f __attribute__((ext_vector_type(2)));
typedef float v8f __attribute__((ext_vector_type(8)));

#define FDIM 128
#define NRM_EPS 1e-12f

// Monotone float -> uint mapping so unsigned atomicMax == float max.
__device__ __forceinline__ unsigned fmap(float f) {
    unsigned u = __float_as_uint(f);
    return (u & 0x80000000u) ? ~u : (u | 0x80000000u);
}
__device__ __forceinline__ float funmap(unsigned u) {
    return __uint_as_float((u & 0x80000000u) ? (u & 0x7fffffffu) : ~u);
}

__global__ void k_init(float* __restrict__ out, unsigned* __restrict__ mmap,
                       float* __restrict__ s, int N, long long ND) {
    long long i = (long long)blockIdx.x * blockDim.x + threadIdx.x;
    if (i < ND) out[i] = 0.0f;
    if (i < N) { mmap[i] = 0u; s[i] = 0.0f; }
}

// One wave per node: inv_norm[n] = 1 / max(||feat[n]||, eps)
__global__ void k_norm(const float* __restrict__ feat, float* __restrict__ inv_norm, int N) {
    int lane = threadIdx.x & 31;
    int node = (int)(((long long)blockIdx.x * blockDim.x + threadIdx.x) >> 5);
    if (node >= N) return;  // wave-uniform
    float4 f = ((const float4*)feat)[(long long)node * 32 + lane];
    float ss = f.x * f.x + f.y * f.y + f.z * f.z + f.w * f.w;
    #pragma unroll
    for (int m = 16; m > 0; m >>= 1) ss += __shfl_xor(ss, m, 32);
    if (lane == 0) inv_norm[node] = 1.0f / fmaxf(sqrtf(ss), NRM_EPS);
}

// One wave per 16 edges: 16x16 Gram tile via V_WMMA_F32_16X16X4_F32 over K=128,
// diagonal = per-edge dot(feat[src], feat[dst]).
__global__ void k_edge_wmma(const float* __restrict__ feat,
                            const int* __restrict__ src,
                            const int* __restrict__ dst,
                            const float* __restrict__ inv_norm,
                            const float* __restrict__ beta,
                            float* __restrict__ e_out,
                            unsigned* __restrict__ mmap,
                            int E) {
    const int lane = threadIdx.x & 31;
    const long long wid = ((long long)blockIdx.x * blockDim.x + threadIdx.x) >> 5;
    const long long base = wid * 16;
    if (base >= E) return;  // wave-uniform

    // A layout (16x4 f32): lanes 0-15 hold M=lane, K={k,k+1}; lanes 16-31 K={k+2,k+3}.
    // B layout (4x16 f32): mirrored -> lane holds col=lane&15, same K split.
    const int m16   = lane & 15;
    const int khalf = (lane >> 4) * 2;
    long long ea = base + m16;
    if (ea > (long long)E - 1) ea = (long long)E - 1;  // tail clamp; EXEC stays full
    const int rowA = src[ea];
    const int rowB = dst[ea];
    const float* pa = feat + (long long)rowA * FDIM + khalf;
    const float* pb = feat + (long long)rowB * FDIM + khalf;

    v8f c0 = {0.f,0.f,0.f,0.f,0.f,0.f,0.f,0.f};
    v8f c1 = {0.f,0.f,0.f,0.f,0.f,0.f,0.f,0.f};
    #pragma unroll
    for (int k = 0; k < 64; k += 4) {
        v2f a0 = *(const v2f*)(pa + k);
        v2f b0 = *(const v2f*)(pb + k);
        c0 = __builtin_amdgcn_wmma_f32_16x16x4_f32(false, a0, false, b0,
                                                   (short)0, c0, false, false);
        v2f a1 = *(const v2f*)(pa + k + 64);
        v2f b1 = *(const v2f*)(pb + k + 64);
        c1 = __builtin_amdgcn_wmma_f32_16x16x4_f32(false, a1, false, b1,
                                                   (short)0, c1, false, false);
    }

    // Diagonal extraction: C[m][m] lives at component m on lane m (m<8)
    // and component m-8 on lane m+16 (m>=8).
    float diag = 0.0f;
    #pragma unroll
    for (int v = 0; v < 8; ++v) {
        float x = c0[v] + c1[v];
        diag = (lane == v)      ? x : diag;
        diag = (lane == v + 24) ? x : diag;
    }

    bool isdiag = (lane < 8) || (lane >= 24);
    long long eidx = base + ((lane < 8) ? lane : (lane - 16));
    if (isdiag && eidx < E) {
        int sI = src[eidx], dI = dst[eidx];
        float ev = beta[0] * diag * inv_norm[sI] * inv_norm[dI];
        e_out[eidx] = ev;
        atomicMax(mmap + dI, fmap(ev));
    }
}

// p = exp(e - m[dst]) (in place over e); s[dst] += p
__global__ void k_expsum(float* __restrict__ ep, const int* __restrict__ dst,
                         const unsigned* __restrict__ mmap, float* __restrict__ s, int E) {
    long long i = (long long)blockIdx.x * blockDim.x + threadIdx.x;
    if (i >= E) return;
    int d = dst[i];
    float pv = __expf(ep[i] - funmap(mmap[d]));
    ep[i] = pv;
    __hip_atomic_fetch_add(s + d, pv, __ATOMIC_RELAXED, __HIP_MEMORY_SCOPE_AGENT);
}

// One wave per edge: out[dst] += feat[src] * (p/s[dst]); lane owns 4 floats.
__global__ void k_scatter(const float* __restrict__ feat, const int* __restrict__ src,
                          const int* __restrict__ dst, const float* __restrict__ p,
                          const float* __restrict__ s, float* __restrict__ out, int E) {
    int lane = threadIdx.x & 31;
    long long edge = ((long long)blockIdx.x * blockDim.x + threadIdx.x) >> 5;
    if (edge >= E) return;  // wave-uniform
    int sI = src[edge], dI = dst[edge];
    float w = p[edge] / s[dI];
    float4 f = ((const float4*)feat)[(long long)sI * 32 + lane];
    float* o = out + (long long)dI * FDIM + lane * 4;
    __hip_atomic_fetch_add(o + 0, f.x * w, __ATOMIC_RELAXED, __HIP_MEMORY_SCOPE_AGENT);
    __hip_atomic_fetch_add(o + 1, f.y * w, __ATOMIC_RELAXED, __HIP_MEMORY_SCOPE_AGENT);
    __hip_atomic_fetch_add(o + 2, f.z * w, __ATOMIC_RELAXED, __HIP_MEMORY_SCOPE_AGENT);
    __hip_atomic_fetch_add(o + 3, f.w * w, __ATOMIC_RELAXED, __HIP_MEMORY_SCOPE_AGENT);
}

extern "C" void kernel_launch(void* const* d_in, const int* in_sizes, int n_in,
                              void* d_out, int out_size, void* d_ws, size_t ws_size,
                              hipStream_t stream) {
    const float* feat = (const float*)d_in[0];
    const float* beta = (const float*)d_in[1];
    const int*   src  = (const int*)d_in[2];
    const int*   dst  = (const int*)d_in[3];
    const int N = in_sizes[0] / FDIM;
    const int E = in_sizes[2];
    float* out = (float*)d_out;

    char* w = (char*)d_ws;
    float*    inv_norm = (float*)w;     w += (size_t)N * sizeof(float);
    unsigned* mmap     = (unsigned*)w;  w += (size_t)N * sizeof(unsigned);
    float*    s        = (float*)w;     w += (size_t)N * sizeof(float);
    float*    ep       = (float*)w;     w += (size_t)E * sizeof(float);  // e, then p in place

    const long long ND = (long long)N * FDIM;
    {
        long long total = ND;  // ND > N, covers both
        int blocks = (int)((total + 255) / 256);
        k_init<<<blocks, 256, 0, stream>>>(out, mmap, s, N, ND);
    }
    {
        int blocks = (N + 7) / 8;  // 8 waves/block, 1 wave/node
        k_norm<<<blocks, 256, 0, stream>>>(feat, inv_norm, N);
    }
    {
        long long waves = ((long long)E + 15) / 16;
        int blocks = (int)((waves + 7) / 8);
        k_edge_wmma<<<blocks, 256, 0, stream>>>(feat, src, dst, inv_norm, beta,
                                                ep, mmap, E);
    }
    {
        int blocks = (E + 255) / 256;
        k_expsum<<<blocks, 256, 0, stream>>>(ep, dst, mmap, s, E);
    }
    {
        long long waves = E;
        int blocks = (int)((waves + 7) / 8);
        k_scatter<<<blocks, 256, 0, stream>>>(feat, src, dst, ep, s, out, E);
    }
}